// Interpolator_42374147342963
// MI455X (gfx1250) — compile-verified
//
#include <hip/hip_runtime.h>

// grid_sample(nearest, padding=border, align_corners=False) with additive
// per-pixel displacement. Pure memory-bound gather:
//   img[B,C,H,W] f32 (96 MiB), x[B,H,W,2] f32 (64 MiB), out[B,C,H,W] (96 MiB)
// Roofline floor ~11.5 us @ 23.3 TB/s; no contraction dim -> WMMA inapplicable.
// Strategy:
//   * 3D grid (row-block, H, B): b,i,j from block indices -> zero divisions.
//   * Fully compile-time-specialized hot kernel <C=3,H=1024,W=1024>:
//     branch-free, 32-bit addressing, literal constants, iy*W -> iy<<10.
//   * NT hints on single-use x/out streams so the 96 MiB img stays in 192MB L2;
//     gathers regular-temporal (L2-resident reuse).
//   * gfx1250 global_prefetch_b8 on the identity line (row i, col j0; displacement
//     ~N(0,1) px, yet gathers hit row ~j, col ~i, so it helps only near i==j).
//   * 4 pixels/thread: b128 x-loads, b128 NT out-stores; 256 thr = 8 wave32.

typedef float v4f __attribute__((ext_vector_type(4)));

template <int CT, int HT, int WT>   // CT>0: full compile-time shape; else runtime
__global__ __launch_bounds__(256) void Interpolator_42374147342963_kernel(
    const float* __restrict__ img,   // [B,C,H,W]
    const float* __restrict__ x,     // [B,H,W,2]
    float* __restrict__ out,         // [B,C,H,W]
    int Cdyn, int Hdyn, int Wdyn)
{
    constexpr bool SPEC = (CT > 0);
    const int C = SPEC ? CT : Cdyn;
    const int H = SPEC ? HT : Hdyn;
    const int W = SPEC ? WT : Wdyn;

    const int b  = blockIdx.z;                                   // uniform
    const int i  = blockIdx.y;                                   // uniform (row)
    const int j0 = (blockIdx.x * blockDim.x + threadIdx.x) * 4;

    const float fW     = (float)W;
    const float fH     = (float)H;
    const float invHm1 = 1.0f / (float)(H - 1);
    const float invWm1 = 1.0f / (float)(W - 1);
    const float maxX   = (float)(W - 1);
    const float maxY   = (float)(H - 1);
    const float g0base = 2.0f * (float)i * invHm1 - 1.0f;        // uniform

    if constexpr (SPEC) {
        // ---- Branch-free specialized path: shapes are compile-time, 32-bit math.
        constexpr int HW   = HT * WT;          // 1M elements
        constexpr int CHW  = CT * HW;          // 3M elements (< 2^31)
        const int rem      = i * WT + j0;      // offset within one image plane
        const int pixBase  = b * HW + rem;     // pixel index (B*HW = 8M < 2^31)

        const float* __restrict__ imgB = img + b * CHW;
        float*       __restrict__ outB = out + b * CHW + rem;

        // Prefetch identity lines (row i, col j0); gathers hit row ~j, col ~i (gfx1250 global_prefetch_b8).
        #pragma unroll
        for (int c = 0; c < CT; ++c)
            __builtin_prefetch(imgB + c * HW + rem, 0, 1);

        // Streamed displacement read: 8 floats = 2x b128, non-temporal.
        const v4f* __restrict__ xv = (const v4f*)(x + (long long)pixBase * 2);
        const v4f x01 = __builtin_nontemporal_load(xv + 0);  // p0{g0,g1} p1{g0,g1}
        const v4f x23 = __builtin_nontemporal_load(xv + 1);  // p2,p3

        const float gx[4] = {x01.x, x01.z, x23.x, x23.z};    // grid ch0 (row-based)
        const float gy[4] = {x01.y, x01.w, x23.y, x23.w};    // grid ch1 (col-based)

        int flat[4];
        #pragma unroll
        for (int p = 0; p < 4; ++p) {
            const float g0 = g0base + 2.0f * gx[p] * invHm1;
            const float g1 = (2.0f * (float)(j0 + p) * invWm1 - 1.0f)
                           + 2.0f * gy[p] * invWm1;
            float fx = ((g0 + 1.0f) * fW - 1.0f) * 0.5f;
            float fy = ((g1 + 1.0f) * fH - 1.0f) * 0.5f;
            fx = fminf(fmaxf(fx, 0.0f), maxX);
            fy = fminf(fmaxf(fy, 0.0f), maxY);
            const int ix = (int)rintf(fx);   // round half-to-even == jnp.round
            const int iy = (int)rintf(fy);
            flat[p] = (iy << 10) + ix;       // WT == 1024
        }

        // Gather (regular-temporal: reuse lives in 192MB L2); NT b128 stores.
        #pragma unroll
        for (int c = 0; c < CT; ++c) {
            const float* __restrict__ ic = imgB + c * HW;
            v4f v;
            v.x = ic[flat[0]];
            v.y = ic[flat[1]];
            v.z = ic[flat[2]];
            v.w = ic[flat[3]];
            __builtin_nontemporal_store(v, (v4f*)(outB + c * HW));
        }
    } else {
        // ---- Generic runtime-shape path.
        if (j0 >= W) return;
        const int       HW     = H * W;
        const long long rowPix = (long long)b * HW + (long long)i * W;
        const int       rem    = i * W + j0;
        const float* __restrict__ imgB = img + (long long)b * C * HW;

        if ((j0 + 3 < W) && (((rowPix + j0) & 3LL) == 0)) {
            const v4f* __restrict__ xv = (const v4f*)(x + (rowPix + j0) * 2LL);
            const v4f x01 = __builtin_nontemporal_load(xv + 0);
            const v4f x23 = __builtin_nontemporal_load(xv + 1);
            const float gx[4] = {x01.x, x01.z, x23.x, x23.z};
            const float gy[4] = {x01.y, x01.w, x23.y, x23.w};

            int flat[4];
            #pragma unroll
            for (int p = 0; p < 4; ++p) {
                const float g0 = g0base + 2.0f * gx[p] * invHm1;
                const float g1 = (2.0f * (float)(j0 + p) * invWm1 - 1.0f)
                               + 2.0f * gy[p] * invWm1;
                float fx = ((g0 + 1.0f) * fW - 1.0f) * 0.5f;
                float fy = ((g1 + 1.0f) * fH - 1.0f) * 0.5f;
                fx = fminf(fmaxf(fx, 0.0f), maxX);
                fy = fminf(fmaxf(fy, 0.0f), maxY);
                flat[p] = (int)rintf(fy) * W + (int)rintf(fx);
            }

            float* __restrict__ outB = out + (long long)b * C * HW + rem;
            for (int c = 0; c < C; ++c) {
                const float* __restrict__ ic = imgB + (long long)c * HW;
                v4f v;
                v.x = ic[flat[0]];
                v.y = ic[flat[1]];
                v.z = ic[flat[2]];
                v.w = ic[flat[3]];
                __builtin_nontemporal_store(v, (v4f*)(outB + (long long)c * HW));
            }
        } else {
            #pragma unroll
            for (int p = 0; p < 4; ++p) {
                const int j = j0 + p;
                if (j >= W) break;
                const float x0 = x[(rowPix + j) * 2 + 0];
                const float x1 = x[(rowPix + j) * 2 + 1];
                const float g0 = g0base + 2.0f * x0 * invHm1;
                const float g1 = (2.0f * (float)j * invWm1 - 1.0f)
                               + 2.0f * x1 * invWm1;
                float fx = ((g0 + 1.0f) * fW - 1.0f) * 0.5f;
                float fy = ((g1 + 1.0f) * fH - 1.0f) * 0.5f;
                fx = fminf(fmaxf(fx, 0.0f), maxX);
                fy = fminf(fmaxf(fy, 0.0f), maxY);
                const int fl = (int)rintf(fy) * W + (int)rintf(fx);

                float* __restrict__ ob = out + (long long)b * C * HW + i * W + j;
                for (int c = 0; c < C; ++c)
                    ob[(long long)c * HW] = imgB[(long long)c * HW + fl];
            }
        }
    }
}

extern "C" void kernel_launch(void* const* d_in, const int* in_sizes, int n_in,
                              void* d_out, int out_size, void* d_ws, size_t ws_size,
                              hipStream_t stream) {
    (void)n_in; (void)out_size; (void)d_ws; (void)ws_size;
    const float* img = (const float*)d_in[0];   // [B,C,H,W] f32
    const float* x   = (const float*)d_in[1];   // [B,H,W,2] f32
    float* out = (float*)d_out;

    // Reference shapes: H = W = 1024; derive B and C from flat sizes.
    const int H = 1024, W = 1024;
    const long long HW = (long long)H * W;
    const int B = (int)((long long)in_sizes[1] / (2LL * HW));          // x = B*H*W*2
    const int C = (int)((long long)in_sizes[0] / ((long long)B * HW)); // img = B*C*H*W

    const int block = 256;                 // 8 wave32 waves, 4 px/thread
    const int pixPerBlock = block * 4;     // 1024 = exactly one row
    dim3 grid((unsigned)((W + pixPerBlock - 1) / pixPerBlock),
              (unsigned)H, (unsigned)B);

    if (C == 3 && H == 1024 && W == 1024) {
        Interpolator_42374147342963_kernel<3, 1024, 1024>
            <<<grid, block, 0, stream>>>(img, x, out, C, H, W);
    } else {
        Interpolator_42374147342963_kernel<0, 0, 0>
            <<<grid, block, 0, stream>>>(img, x, out, C, H, W);
    }
}